// MACENodeMessageBlock_40724879901208
// MI455X (gfx1250) — compile-verified
//
#include <hip/hip_runtime.h>
#include <hip/hip_bf16.h>
#include <math.h>

// ---------------------------------------------------------------------------
// MACE node message block for gfx1250 (MI455X), full fp32 via
// V_WMMA_F32_16X16X4_F32. Pipeline:
//   K1: node up-projection  (s_up, v_up)            -- WMMA f32
//   K2: fused edge MLP -> tensor-product messages -> atomic segment_sum
//   K3: output linears -> d_out                      -- WMMA f32
// ---------------------------------------------------------------------------

typedef __attribute__((ext_vector_type(2))) float v2f;
typedef __attribute__((ext_vector_type(8))) float v8f;

#define CC    128     // C
#define RBD   8       // RB
#define HIDD  64      // HID
#define HPAD  66      // LDS row pitch (bank-spread, keeps 8B alignment)
#define EW    4       // waves per block in edge kernel

static __device__ __forceinline__ v8f wmma4(v2f a, v2f b, v8f c) {
  // D = A(16x4 f32) * B(4x16 f32) + C(16x16 f32)
  return __builtin_amdgcn_wmma_f32_16x16x4_f32(false, a, false, b,
                                               (short)0, c, false, false);
}

static __device__ __forceinline__ float silu_f(float x) {
  return x * (1.0f / (1.0f + __expf(-x)));
}

// A fragment from row-major X (pitch ldx): lanes 0-15 -> K=k0..k0+1,
// lanes 16-31 -> K=k0+2..k0+3 (ISA 7.12.2 "32-bit A-Matrix 16x4").
static __device__ __forceinline__ v2f load_a_g(const float* __restrict__ X,
                                               int ldx, int r0, int k0, int lane) {
  const int m    = lane & 15;
  const int koff = (lane >> 4) << 1;
  const float* p = X + (size_t)(r0 + m) * ldx + k0 + koff;
  v2f a; a.x = p[0]; a.y = p[1];
  return a;
}

// B fragment from row-major W (K x Ncols, pitch ldw), mirrored K striping.
static __device__ __forceinline__ v2f load_b_g(const float* __restrict__ W,
                                               int ldw, int k0, int n0, int lane) {
  const int n    = lane & 15;
  const int koff = (lane >> 4) << 1;
  v2f b;
  b.x = W[(size_t)(k0 + koff)     * ldw + n0 + n];
  b.y = W[(size_t)(k0 + koff + 1) * ldw + n0 + n];
  return b;
}

// A fragment from an LDS 16xHIDD activation tile (pitch HPAD).
static __device__ __forceinline__ v2f load_a_lds(const float* h, int k0, int lane) {
  const int m    = lane & 15;
  const int koff = (lane >> 4) << 1;
  const float* p = h + m * HPAD + k0 + koff;
  v2f a; a.x = p[0]; a.y = p[1];
  return a;
}

// D tile -> LDS with scale (+ optional SiLU). VGPR r holds rows r / r+8.
static __device__ __forceinline__ void store_d_lds(float* h, int n0, v8f d,
                                                   float scale, bool act, int lane) {
  const int col   = n0 + (lane & 15);
  const int rbase = (lane >> 4) << 3;
#pragma unroll
  for (int r = 0; r < 8; ++r) {
    float x = d[r] * scale;
    if (act) x = silu_f(x);
    h[(rbase + r) * HPAD + col] = x;
  }
}

// ---------------------------------------------------------------------------
// K1: s_up = s @ W_up_s/sqrt(C);  v_up[n,d,m] = sum_c v[n,c,m] W_up_v[c,d]/sqrt(C)
// grid = (N/16, C/16, 4)  z: 0=scalar, 1..3 = vector component
// ---------------------------------------------------------------------------
__global__ void __launch_bounds__(32)
mace_node_up(const float* __restrict__ node_feats,
             const float* __restrict__ W_up_s,
             const float* __restrict__ W_up_v,
             float* __restrict__ s_up, float* __restrict__ v_up, int Nn) {
  const int lane = threadIdx.x;
  const int r0   = blockIdx.x * 16;
  const int n0   = blockIdx.y * 16;
  const int comp = blockIdx.z;
  if (r0 >= Nn) return;
  const float scale = 0.08838834764831845f; // 1/sqrt(128)
  const int col   = n0 + (lane & 15);
  const int rbase = (lane >> 4) << 3;
  v8f acc = {};
  if (comp == 0) {
    for (int k0 = 0; k0 < CC; k0 += 4) {
      v2f a = load_a_g(node_feats, 4 * CC, r0, k0, lane);
      v2f b = load_b_g(W_up_s, CC, k0, n0, lane);
      acc = wmma4(a, b, acc);
    }
#pragma unroll
    for (int r = 0; r < 8; ++r)
      s_up[(size_t)(r0 + rbase + r) * CC + col] = acc[r] * scale;
  } else {
    const int mc   = comp - 1;
    const int m    = lane & 15;
    const int koff = (lane >> 4) << 1;
    for (int k0 = 0; k0 < CC; k0 += 4) {
      const float* p = node_feats + (size_t)(r0 + m) * (4 * CC) + CC
                     + (size_t)(k0 + koff) * 3 + mc;
      v2f a; a.x = p[0]; a.y = p[3];   // stride-3 channel layout
      v2f b = load_b_g(W_up_v, CC, k0, n0, lane);
      acc = wmma4(a, b, acc);
    }
#pragma unroll
    for (int r = 0; r < 8; ++r)
      v_up[((size_t)(r0 + rbase + r) * CC + col) * 3 + mc] = acc[r] * scale;
  }
}

// ---------------------------------------------------------------------------
// K2: fused edge MLP (3x SiLU layers + final 64->512) + tensor-product
// messages + atomic segment_sum. One wave per 16-edge tile, EW waves/block.
// ---------------------------------------------------------------------------
__global__ void __launch_bounds__(EW * 32)
mace_edge(const float* __restrict__ edge_attrs,
          const float* __restrict__ edge_feats,
          const int*   __restrict__ edge_index,
          const float* __restrict__ W1, const float* __restrict__ W2,
          const float* __restrict__ W3, const float* __restrict__ W4,
          const float* __restrict__ s_up, const float* __restrict__ v_up,
          float* __restrict__ msg_s, float* __restrict__ msg_v, int Ee) {
  __shared__ float hA[EW][16][HPAD];
  __shared__ float hB[EW][16][HPAD];
  __shared__ float ey[EW][16][4];   // y0, y1x, y1y, y1z
  __shared__ int   esd[EW][16];     // sender
  __shared__ int   erc[EW][16];     // receiver

  const int wave = threadIdx.x >> 5;
  const int lane = threadIdx.x & 31;
  const int e0   = (blockIdx.x * EW + wave) * 16;
  if (e0 >= Ee) return;

  if (lane < 16) {
    const int e = e0 + lane;
    esd[wave][lane] = edge_index[e];
    erc[wave][lane] = edge_index[Ee + e];
    ey[wave][lane][0] = edge_attrs[(size_t)e * 4 + 0];
    ey[wave][lane][1] = edge_attrs[(size_t)e * 4 + 1];
    ey[wave][lane][2] = edge_attrs[(size_t)e * 4 + 2];
    ey[wave][lane][3] = edge_attrs[(size_t)e * 4 + 3];
  }

  // Layer 1: (16x8) @ W1(8x64) * 1/sqrt(8), SiLU -> hA
  for (int ct = 0; ct < 4; ++ct) {
    v8f acc = {};
    for (int k0 = 0; k0 < RBD; k0 += 4) {
      v2f a = load_a_g(edge_feats, RBD, e0, k0, lane);
      v2f b = load_b_g(W1, HIDD, k0, ct * 16, lane);
      acc = wmma4(a, b, acc);
    }
    store_d_lds(&hA[wave][0][0], ct * 16, acc, 0.35355339059327373f, true, lane);
  }
  // Layer 2: hA @ W2 * 1/8, SiLU -> hB
  for (int ct = 0; ct < 4; ++ct) {
    v8f acc = {};
    for (int k0 = 0; k0 < HIDD; k0 += 4) {
      v2f a = load_a_lds(&hA[wave][0][0], k0, lane);
      v2f b = load_b_g(W2, HIDD, k0, ct * 16, lane);
      acc = wmma4(a, b, acc);
    }
    store_d_lds(&hB[wave][0][0], ct * 16, acc, 0.125f, true, lane);
  }
  // Layer 3: hB @ W3 * 1/8, SiLU -> hA
  for (int ct = 0; ct < 4; ++ct) {
    v8f acc = {};
    for (int k0 = 0; k0 < HIDD; k0 += 4) {
      v2f a = load_a_lds(&hB[wave][0][0], k0, lane);
      v2f b = load_b_g(W3, HIDD, k0, ct * 16, lane);
      acc = wmma4(a, b, acc);
    }
    store_d_lds(&hA[wave][0][0], ct * 16, acc, 0.125f, true, lane);
  }

  // Layer 4 fused with message formation: per 16-channel tile compute the
  // four tpw[:,j,:] column tiles in registers, then gather/compute/scatter.
  const float C000 = 0.70710678118654752f;  // sqrt(0.5)
  const float C110 = 0.40824829046386302f;  // sqrt(0.5)/sqrt(3)
  const float C011 = 0.70710678118654752f;  // sqrt(1.5)/sqrt(3)
  const float C101 = 0.70710678118654752f;
  const int cl    = lane & 15;
  const int rbase = (lane >> 4) << 3;

  for (int ct = 0; ct < 8; ++ct) {
    const int c0 = ct * 16;
    v8f d0 = {}, d1 = {}, d2 = {}, d3 = {};
    for (int k0 = 0; k0 < HIDD; k0 += 4) {
      v2f a  = load_a_lds(&hA[wave][0][0], k0, lane);
      v2f b0 = load_b_g(W4, 4 * CC, k0, 0 * CC + c0, lane);
      v2f b1 = load_b_g(W4, 4 * CC, k0, 1 * CC + c0, lane);
      v2f b2 = load_b_g(W4, 4 * CC, k0, 2 * CC + c0, lane);
      v2f b3 = load_b_g(W4, 4 * CC, k0, 3 * CC + c0, lane);
      d0 = wmma4(a, b0, d0);
      d1 = wmma4(a, b1, d1);
      d2 = wmma4(a, b2, d2);
      d3 = wmma4(a, b3, d3);
    }
    const int ch = c0 + cl;
#pragma unroll
    for (int r = 0; r < 8; ++r) {
      const int er  = rbase + r;
      const int snd = esd[wave][er];
      const int rcv = erc[wave][er];
      const float y0  = ey[wave][er][0];
      const float y1x = ey[wave][er][1];
      const float y1y = ey[wave][er][2];
      const float y1z = ey[wave][er][3];
      const float t0 = d0[r] * 0.125f;   // fold 1/sqrt(HID)
      const float t1 = d1[r] * 0.125f;
      const float t2 = d2[r] * 0.125f;
      const float t3 = d3[r] * 0.125f;
      const float se = s_up[(size_t)snd * CC + ch];
      const float* vp = v_up + ((size_t)snd * CC + ch) * 3;
      const float v0 = vp[0], v1 = vp[1], v2 = vp[2];
      float* ms = msg_s + (size_t)rcv * (2 * CC);
      float* mv = msg_v + (size_t)rcv * (2 * CC) * 3;
      atomicAdd(ms + ch,      C000 * t0 * se * y0);
      atomicAdd(ms + CC + ch, C110 * t3 * (v0 * y1x + v1 * y1y + v2 * y1z));
      const float s1 = C011 * t1 * se;
      atomicAdd(mv + (size_t)ch * 3 + 0, s1 * y1x);
      atomicAdd(mv + (size_t)ch * 3 + 1, s1 * y1y);
      atomicAdd(mv + (size_t)ch * 3 + 2, s1 * y1z);
      const float s2 = C101 * t2 * y0;
      atomicAdd(mv + (size_t)(CC + ch) * 3 + 0, s2 * v0);
      atomicAdd(mv + (size_t)(CC + ch) * 3 + 1, s2 * v1);
      atomicAdd(mv + (size_t)(CC + ch) * 3 + 2, s2 * v2);
    }
  }
}

// ---------------------------------------------------------------------------
// K3: out_s = msg_s @ W_lin_s/16 /16 ; out_v = einsum(msg_v, W_lin_v)/16 /16
// out[n] = [out_s (128) | out_v interleaved (c*3+m, 384)]
// grid = (N/16, C/16, 4)  z: 0=scalar, 1..3 = vector component
// ---------------------------------------------------------------------------
__global__ void __launch_bounds__(32)
mace_out(const float* __restrict__ msg_s, const float* __restrict__ msg_v,
         const float* __restrict__ W_lin_s, const float* __restrict__ W_lin_v,
         float* __restrict__ out, int Nn) {
  const int lane = threadIdx.x;
  const int r0   = blockIdx.x * 16;
  const int n0   = blockIdx.y * 16;
  const int comp = blockIdx.z;
  if (r0 >= Nn) return;
  const float scale = 0.00390625f; // (1/sqrt(2C)) * (1/AVG_NEIGH) = (1/16)/16
  const int col   = n0 + (lane & 15);
  const int rbase = (lane >> 4) << 3;
  v8f acc = {};
  if (comp == 0) {
    for (int k0 = 0; k0 < 2 * CC; k0 += 4) {
      v2f a = load_a_g(msg_s, 2 * CC, r0, k0, lane);
      v2f b = load_b_g(W_lin_s, CC, k0, n0, lane);
      acc = wmma4(a, b, acc);
    }
#pragma unroll
    for (int r = 0; r < 8; ++r)
      out[(size_t)(r0 + rbase + r) * 512 + col] = acc[r] * scale;
  } else {
    const int mc   = comp - 1;
    const int m    = lane & 15;
    const int koff = (lane >> 4) << 1;
    for (int k0 = 0; k0 < 2 * CC; k0 += 4) {
      const float* p = msg_v + ((size_t)(r0 + m) * (2 * CC) + k0 + koff) * 3 + mc;
      v2f a; a.x = p[0]; a.y = p[3];
      v2f b = load_b_g(W_lin_v, CC, k0, n0, lane);
      acc = wmma4(a, b, acc);
    }
#pragma unroll
    for (int r = 0; r < 8; ++r)
      out[(size_t)(r0 + rbase + r) * 512 + CC + col * 3 + mc] = acc[r] * scale;
  }
}

// ---------------------------------------------------------------------------
extern "C" void kernel_launch(void* const* d_in, const int* in_sizes, int n_in,
                              void* d_out, int out_size, void* d_ws, size_t ws_size,
                              hipStream_t stream) {
  // setup_inputs() order
  const float* node_feats = (const float*)d_in[1];
  const float* edge_attrs = (const float*)d_in[2];
  const float* edge_feats = (const float*)d_in[3];
  const int*   edge_index = (const int*)  d_in[4];
  const float* W_up_s  = (const float*)d_in[5];
  const float* W_up_v  = (const float*)d_in[6];
  const float* W_mlp1  = (const float*)d_in[7];
  const float* W_mlp2  = (const float*)d_in[8];
  const float* W_mlp3  = (const float*)d_in[9];
  const float* W_mlp4  = (const float*)d_in[10];
  const float* W_lin_s = (const float*)d_in[11];
  const float* W_lin_v = (const float*)d_in[12];

  const int Nn = in_sizes[1] / (4 * CC);   // 10000
  const int Ee = in_sizes[3] / RBD;        // 160000

  // Workspace layout (floats): s_up | v_up | msg_s | msg_v
  float* s_up  = (float*)d_ws;
  float* v_up  = s_up  + (size_t)Nn * CC;
  float* msg_s = v_up  + (size_t)Nn * CC * 3;
  float* msg_v = msg_s + (size_t)Nn * 2 * CC;

  // Zero accumulation buffers (graph-capture-safe async memset).
  hipMemsetAsync(msg_s, 0, sizeof(float) * (size_t)Nn * 8 * CC, stream);

  const int rtiles = (Nn + 15) / 16;           // 625
  mace_node_up<<<dim3(rtiles, CC / 16, 4), 32, 0, stream>>>(
      node_feats, W_up_s, W_up_v, s_up, v_up, Nn);

  const int etiles = (Ee + 16 * EW - 1) / (16 * EW);  // 2500
  mace_edge<<<dim3(etiles), EW * 32, 0, stream>>>(
      edge_attrs, edge_feats, edge_index, W_mlp1, W_mlp2, W_mlp3, W_mlp4,
      s_up, v_up, msg_s, msg_v, Ee);

  mace_out<<<dim3(rtiles, CC / 16, 4), 32, 0, stream>>>(
      msg_s, msg_v, W_lin_s, W_lin_v, (float*)d_out, Nn);
}